// CascadeCore_3882650436478
// MI455X (gfx1250) — compile-verified
//
#include <hip/hip_runtime.h>
#include <math.h>

// ---------------------------------------------------------------------------
// CDNA5 / gfx1250 implementation. Matrix math uses native fp32 WMMA
// (V_WMMA_F32_16X16X4_F32) to keep reference fp32 precision on the matrix
// pipes. Workload is tiny (~4.9 GFLOP, ~35MB, fits in 192MB L2) -> short chain
// of dense WMMA kernels, one 16x16 output tile per wave32.
// ---------------------------------------------------------------------------

typedef __attribute__((ext_vector_type(2))) float v2f;
typedef __attribute__((ext_vector_type(8))) float v8f;

#define BQ   32
#define SQ   16
#define RQ   128
#define RNNQ 1024
#define FGN  461
#define MINV (-1e8f)

__device__ __forceinline__ v8f wmma_f32(v2f a, v2f b, v8f c) {
  // (neg_a, A, neg_b, B, c_mod, C, reuse_a, reuse_b)
  return __builtin_amdgcn_wmma_f32_16x16x4_f32(false, a, false, b, (short)0, c,
                                               false, false);
}

// ---------------------------------------------------------------------------
// 1 / max(1, sum_r roi_labels[b,s,r])
// ---------------------------------------------------------------------------
__global__ void k_cnt(const float* __restrict__ roi, float* __restrict__ cnt_inv) {
  int i = blockIdx.x * blockDim.x + threadIdx.x;
  if (i >= BQ * SQ) return;
  const float* p = roi + (size_t)i * RQ;
  float s = 0.f;
  #pragma unroll 8
  for (int r = 0; r < RQ; ++r) s += p[r];
  cnt_inv[i] = 1.0f / (s == 0.f ? 1.0f : s);
}

// ---------------------------------------------------------------------------
// pooled[b,s,f] = (1/cnt[b,s]) * sum_r roi[b,s,r] * feats[b,r,f]
// Per batch: (16 x 128) @ (128 x 1024). One 16x16 N-tile per wave.
// grid: 512 blocks x 128 thr (4 waves) -> 2048 tiles = 32 b * 64 ntiles
// ---------------------------------------------------------------------------
__global__ void k_pool(const float* __restrict__ roi,
                       const float* __restrict__ feats,
                       const float* __restrict__ cnt_inv,
                       float* __restrict__ pooled) {
  int wid = blockIdx.x * (blockDim.x >> 5) + (threadIdx.x >> 5);
  int b  = wid >> 6;
  int nt = wid & 63;
  int n_base = nt * 16;
  int l    = threadIdx.x & 31;
  int lr   = l & 15;
  int koff = (l >> 4) << 1;               // 0 or 2
  const float* A  = roi   + (size_t)b * SQ * RQ;    // 16 x 128 row-major
  const float* Bm = feats + (size_t)b * RQ * RNNQ;  // 128 x 1024 row-major
  const float* Arow = A + (size_t)lr * RQ;
  int n = n_base + lr;
  v8f acc = {};
  #pragma unroll 4
  for (int k = 0; k < RQ; k += 4) {
    v2f a, bb;
    a.x  = Arow[k + koff];
    a.y  = Arow[k + koff + 1];
    bb.x = Bm[(size_t)(k + koff)     * RNNQ + n];
    bb.y = Bm[(size_t)(k + koff + 1) * RNNQ + n];
    acc = wmma_f32(a, bb, acc);
  }
  int mhalf = (l >> 4) << 3;              // 0 or 8
  #pragma unroll
  for (int v = 0; v < 8; ++v) {
    int row = b * SQ + v + mhalf;
    pooled[(size_t)row * RNNQ + n] = acc[v] * cnt_inv[row];
  }
}

// ---------------------------------------------------------------------------
// H = relu([rnn | pooled](512x2048) @ W^T + bias),  W: (1024 x 2048) row-major
// Two K-phases (rnn cols 0..1023 with W[:, :1024], pooled with W[:, 1024:]).
// grid: 512 blocks x 128 thr -> 2048 tiles = 32 mtiles * 64 ntiles
// ---------------------------------------------------------------------------
__global__ void k_gemm_2048_relu(const float* __restrict__ rnn,
                                 const float* __restrict__ pooled,
                                 const float* __restrict__ W,
                                 const float* __restrict__ bias,
                                 float* __restrict__ H) {
  int wid = blockIdx.x * (blockDim.x >> 5) + (threadIdx.x >> 5);
  int mt = wid >> 6;
  int nt = wid & 63;
  int m_base = mt * 16, n_base = nt * 16;
  int l    = threadIdx.x & 31;
  int lr   = l & 15;
  int koff = (l >> 4) << 1;
  const float* Arow0 = rnn    + (size_t)(m_base + lr) * RNNQ;
  const float* Arow1 = pooled + (size_t)(m_base + lr) * RNNQ;
  const float* Wrow  = W      + (size_t)(n_base + lr) * (2 * RNNQ);
  v8f acc = {};
  #pragma unroll 4
  for (int k = 0; k < RNNQ; k += 4) {
    v2f a, bb;
    a.x  = Arow0[k + koff];        a.y  = Arow0[k + koff + 1];
    bb.x = Wrow [k + koff];        bb.y = Wrow [k + koff + 1];
    acc = wmma_f32(a, bb, acc);
  }
  #pragma unroll 4
  for (int k = 0; k < RNNQ; k += 4) {
    v2f a, bb;
    a.x  = Arow1[k + koff];            a.y  = Arow1[k + koff + 1];
    bb.x = Wrow [RNNQ + k + koff];     bb.y = Wrow [RNNQ + k + koff + 1];
    acc = wmma_f32(a, bb, acc);
  }
  int n = n_base + lr;
  int mhalf = (l >> 4) << 3;
  float bn = bias[n];
  #pragma unroll
  for (int v = 0; v < 8; ++v) {
    int m = m_base + v + mhalf;
    float val = acc[v] + bn;
    H[(size_t)m * RNNQ + n] = val > 0.f ? val : 0.f;
  }
}

// ---------------------------------------------------------------------------
// fg_out(512x300, stride 304) = h_fg(512x1024) @ W_fg2^T + b,  W_fg2:(300x1024)
// 32 mtiles * 19 ntiles = 608 waves -> 152 blocks x 128 thr
// ---------------------------------------------------------------------------
__global__ void k_gemm_fg2(const float* __restrict__ Hfg,
                           const float* __restrict__ W,
                           const float* __restrict__ bias,
                           float* __restrict__ out) {
  int wid = blockIdx.x * (blockDim.x >> 5) + (threadIdx.x >> 5);
  if (wid >= 32 * 19) return;
  int mt = wid / 19, nt = wid % 19;
  int m_base = mt * 16, n_base = nt * 16;
  int l    = threadIdx.x & 31;
  int lr   = l & 15;
  int koff = (l >> 4) << 1;
  int n = n_base + lr;
  int nclamp = n < 300 ? n : 299;        // clamp OOB weight rows (guarded store)
  const float* Arow = Hfg + (size_t)(m_base + lr) * RNNQ;
  const float* Wrow = W   + (size_t)nclamp * RNNQ;
  v8f acc = {};
  #pragma unroll 4
  for (int k = 0; k < RNNQ; k += 4) {
    v2f a, bb;
    a.x  = Arow[k + koff];  a.y  = Arow[k + koff + 1];
    bb.x = Wrow[k + koff];  bb.y = Wrow[k + koff + 1];
    acc = wmma_f32(a, bb, acc);
  }
  if (n < 300) {
    int mhalf = (l >> 4) << 3;
    float bn = bias[n];
    #pragma unroll
    for (int v = 0; v < 8; ++v) {
      int m = m_base + v + mhalf;
      out[(size_t)m * 304 + n] = acc[v] + bn;
    }
  }
}

// ---------------------------------------------------------------------------
// fg_score(512x461, stride 464) = fg_out(512x300) @ fg_emb^T, masked to MINV
// fg_emb: (461 x 300) row-major. fg_mask: raw jax bool bytes (1B/elem).
// 32 mtiles * 29 ntiles = 928 waves -> 232 blocks x 128 thr
// ---------------------------------------------------------------------------
__global__ void k_gemm_score(const float* __restrict__ fgout,
                             const float* __restrict__ emb,
                             const int* __restrict__ fg_idx,
                             const unsigned char* __restrict__ mask,
                             float* __restrict__ score) {
  int wid = blockIdx.x * (blockDim.x >> 5) + (threadIdx.x >> 5);
  if (wid >= 32 * 29) return;
  int mt = wid / 29, nt = wid % 29;
  int m_base = mt * 16, n_base = nt * 16;
  int l    = threadIdx.x & 31;
  int lr   = l & 15;
  int koff = (l >> 4) << 1;
  int n = n_base + lr;
  int nclamp = n < FGN ? n : (FGN - 1);
  const float* Arow = fgout + (size_t)(m_base + lr) * 304;
  const float* Erow = emb   + (size_t)nclamp * 300;
  v8f acc = {};
  #pragma unroll 4
  for (int k = 0; k < 300; k += 4) {
    v2f a, bb;
    a.x  = Arow[k + koff];  a.y  = Arow[k + koff + 1];
    bb.x = Erow[k + koff];  bb.y = Erow[k + koff + 1];
    acc = wmma_f32(a, bb, acc);
  }
  if (n < FGN) {
    int mhalf = (l >> 4) << 3;
    #pragma unroll
    for (int v = 0; v < 8; ++v) {
      int m = m_base + v + mhalf;
      int idx = fg_idx[m];
      float val = acc[v];
      if (mask[(size_t)idx * FGN + n]) val = MINV;
      score[(size_t)m * 464 + n] = val;
    }
  }
}

// ---------------------------------------------------------------------------
// bn head: per row, z = h_bn(1024) @ W_bn2^T(1024x2) + b; 2-way log_softmax.
// One wave32 per row; 64 blocks x 256 thr (8 waves).
// ---------------------------------------------------------------------------
__global__ void k_bn_head(const float* __restrict__ Hbn,
                          const float* __restrict__ W2,
                          const float* __restrict__ b2,
                          float* __restrict__ out) {
  int row = blockIdx.x * (blockDim.x >> 5) + (threadIdx.x >> 5);
  if (row >= BQ * SQ) return;
  int l = threadIdx.x & 31;
  const float* h = Hbn + (size_t)row * RNNQ;
  float a0 = 0.f, a1 = 0.f;
  for (int j = l; j < RNNQ; j += 32) {
    float hv = h[j];
    a0 += hv * W2[j];
    a1 += hv * W2[RNNQ + j];
  }
  #pragma unroll
  for (int off = 16; off > 0; off >>= 1) {
    a0 += __shfl_xor(a0, off, 32);
    a1 += __shfl_xor(a1, off, 32);
  }
  if (l == 0) {
    float z0 = a0 + b2[0], z1 = a1 + b2[1];
    float mx = fmaxf(z0, z1);
    float lse = mx + logf(expf(z0 - mx) + expf(z1 - mx));
    out[row * 2 + 0] = z0 - lse;
    out[row * 2 + 1] = z1 - lse;
  }
}

// ---------------------------------------------------------------------------
// fg log_softmax over 461 per row. One block (256 thr) per row, 512 blocks.
// ---------------------------------------------------------------------------
__global__ void k_fg_softmax(const float* __restrict__ score,
                             float* __restrict__ out) {
  __shared__ float red[256];
  int row = blockIdx.x, t = threadIdx.x;
  const float* s = score + (size_t)row * 464;
  float mx = -3.4e38f;
  for (int j = t; j < FGN; j += 256) mx = fmaxf(mx, s[j]);
  red[t] = mx; __syncthreads();
  for (int st = 128; st > 0; st >>= 1) {
    if (t < st) red[t] = fmaxf(red[t], red[t + st]);
    __syncthreads();
  }
  mx = red[0]; __syncthreads();
  float sum = 0.f;
  for (int j = t; j < FGN; j += 256) sum += expf(s[j] - mx);
  red[t] = sum; __syncthreads();
  for (int st = 128; st > 0; st >>= 1) {
    if (t < st) red[t] += red[t + st];
    __syncthreads();
  }
  float lse = mx + logf(red[0]);
  float* o = out + (size_t)(BQ * SQ) * 2 + (size_t)row * FGN;
  for (int j = t; j < FGN; j += 256) o[j] = s[j] - lse;
}

// ---------------------------------------------------------------------------
extern "C" void kernel_launch(void* const* d_in, const int* in_sizes, int n_in,
                              void* d_out, int out_size, void* d_ws, size_t ws_size,
                              hipStream_t stream) {
  const int*   fg_idx     = (const int*)d_in[0];
  const float* pool_feats = (const float*)d_in[1];
  const float* rnn_outs   = (const float*)d_in[2];
  const float* roi        = (const float*)d_in[3];
  // d_in[4], d_in[5]: seq_batch_size / seq_cnt (compile-time constants here)
  const float* W_bn1 = (const float*)d_in[6];
  const float* b_bn1 = (const float*)d_in[7];
  const float* W_bn2 = (const float*)d_in[8];
  const float* b_bn2 = (const float*)d_in[9];
  const float* W_fg1 = (const float*)d_in[10];
  const float* b_fg1 = (const float*)d_in[11];
  const float* W_fg2 = (const float*)d_in[12];
  const float* b_fg2 = (const float*)d_in[13];
  const float* fg_emb = (const float*)d_in[14];
  const unsigned char* fg_mask = (const unsigned char*)d_in[15]; // jax bool = 1 byte

  float* ws       = (float*)d_ws;
  float* cnt_inv  = ws;                         // 512
  float* pooled   = cnt_inv + 512;              // 512*1024
  float* h_bn     = pooled  + 512 * 1024;       // 512*1024
  float* h_fg     = h_bn    + 512 * 1024;       // 512*1024
  float* fg_out   = h_fg    + 512 * 1024;       // 512*304
  float* fg_score = fg_out  + 512 * 304;        // 512*464
  float* out      = (float*)d_out;              // [512*2 | 512*461]

  k_cnt          <<<  2, 256, 0, stream>>>(roi, cnt_inv);
  k_pool         <<<512, 128, 0, stream>>>(roi, pool_feats, cnt_inv, pooled);
  k_gemm_2048_relu<<<512, 128, 0, stream>>>(rnn_outs, pooled, W_bn1, b_bn1, h_bn);
  k_gemm_2048_relu<<<512, 128, 0, stream>>>(rnn_outs, pooled, W_fg1, b_fg1, h_fg);
  k_bn_head      <<< 64, 256, 0, stream>>>(h_bn, W_bn2, b_bn2, out);
  k_gemm_fg2     <<<152, 128, 0, stream>>>(h_fg, W_fg2, b_fg2, fg_out);
  k_gemm_score   <<<232, 128, 0, stream>>>(fg_out, fg_emb, fg_idx, fg_mask, fg_score);
  k_fg_softmax   <<<512, 256, 0, stream>>>(fg_score, out);
}